// RJAN_57621281243407
// MI455X (gfx1250) — compile-verified
//
#include <hip/hip_runtime.h>
#include <math.h>

// ---------------------------------------------------------------------------
// Problem constants (RGA module, B=64, C=128, H=W=28)
// ---------------------------------------------------------------------------
#define BB 64
#define CC 128
#define SS 784      // H*W
#define ICN 32      // inter_channel
#define ISN 196     // inter_spatial
#define DSN 49      // spatial bottleneck
#define DCN 8       // channel bottleneck

#define WAVES 4     // waves per block in the GEMM kernel (wave32)
#define KCH  64     // K-chunk staged in LDS
#define KPAD 66     // padded LDS row stride (floats): 8B aligned, bank-spread

typedef float v2f __attribute__((ext_vector_type(2)));
typedef float v8f __attribute__((ext_vector_type(8)));

__device__ __forceinline__ v8f wmma_f32(v2f a, v2f b, v8f c) {
    return __builtin_amdgcn_wmma_f32_16x16x4_f32(false, a, false, b,
                                                 (short)0, c, false, false);
}

// CDNA5 async global->LDS copy (ASYNCcnt) ; LDS byte addr = low 32 bits of the
// generic shared-pointer (flat LDS addresses truncate to the DS offset).
__device__ __forceinline__ void async_g2lds_b32(float* lds_p, const float* g_p) {
    const unsigned lds_off = (unsigned)(unsigned long long)(uintptr_t)lds_p;
    asm volatile("global_load_async_to_lds_b32 %0, %1, off"
                 :: "v"(lds_off), "v"(g_p) : "memory");
}
__device__ __forceinline__ void wait_async() {
    asm volatile("s_wait_asynccnt 0x0" ::: "memory");
}

// ---------------------------------------------------------------------------
// Batched f32 WMMA GEMM:  C[b] (M x N) = op(A[b]) * op(B[b])   [+C] [affine+relu]
//   TA: A stored [K x M] (A[k*lda+m]) else [M x K] (A[m*lda+k])
//   TB: B stored [N x K] (B[n*ldb+k]) else [K x N] (B[k*ldb+n])
// Block = 4 waves sharing one 16-row A tile; each wave owns a 16x32 C tile.
// A tile is double-buffered in LDS, filled with global_load_async_to_lds_b32
// (coalesced, zero-padded past K so full chunks need no masking), consumed as
// ds_load_b64 fragments. B is read with uniform-base + 32-bit lane offsets
// (saddr-form loads). 2 WMMAs per 4-k step; safe chunks fully unrolled.
// ---------------------------------------------------------------------------
template<int TA, int TB>
__global__ __launch_bounds__(WAVES * 32)
void rga_wmma_gemm_f32(const float* __restrict__ A, const float* __restrict__ B,
                       float* __restrict__ C,
                       int M, int N, int K, int lda, int ldb, int ldc,
                       long long sA, long long sB, long long sC,
                       const float* __restrict__ scale,
                       const float* __restrict__ shift,
                       int relu, int accum)
{
    __shared__ float lsA[2][16 * KPAD];

    const int tid  = threadIdx.x;
    const int lane = tid & 31;
    const int wave = tid >> 5;
    const int m0   = blockIdx.y * 16;
    const int n0   = (blockIdx.x * WAVES + wave) * 32;
    const int b    = blockIdx.z;
    const bool active = (n0 < N);                 // wave-uniform

    A += (long long)b * sA;
    B += (long long)b * sB;
    C += (long long)b * sC;

    const int ll = lane & 15;                     // lane within half-wave
    const int lh = lane >> 4;                     // half-wave k-pair select
    const int bn0 = min(n0 + ll, N - 1);          // clamped B cols
    const int bn1 = min(n0 + 16 + ll, N - 1);
    const int bOff = TB ? 1 : ldb;                // b1 relative to b0

    // Cooperative async fill of A chunk [16 x KCH] (zero pad for k >= K).
    auto fill = [&](int buf, int kc) {
        const int len = min(KCH, K - kc);
        #pragma unroll
        for (int i = 0; i < (16 * KCH) / (WAVES * 32); ++i) {
            const int idx = tid + i * (WAVES * 32);
            int r, j;
            if (TA) { r = idx & 15; j = idx >> 4; }   // coalesced over m
            else    { j = idx & 63; r = idx >> 6; }   // coalesced over k
            const int rm = min(m0 + r, M - 1);
            float* lp = &lsA[buf][r * KPAD + j];
            if (j < len) {
                const float* gp = TA ? (A + (long long)(kc + j) * lda + rm)
                                     : (A + (long long)rm * lda + (kc + j));
                async_g2lds_b32(lp, gp);
            } else {
                *lp = 0.f;                             // zero pad (K tail)
            }
        }
    };

    v8f acc0 = {}, acc1 = {};

    // Full chunk: compile-time 16 steps, unguarded.
    auto compute_full = [&](int buf, int kc) {
        const float* lp = &lsA[buf][ll * KPAD + 2 * lh];
        int oB0 = TB ? (bn0 * ldb + kc + 2 * lh) : ((kc + 2 * lh) * ldb + bn0);
        int oB1 = TB ? (bn1 * ldb + kc + 2 * lh) : ((kc + 2 * lh) * ldb + bn1);
        const int bStep = TB ? 4 : 4 * ldb;
        #pragma unroll
        for (int k = 0; k < KCH; k += 4) {
            v2f av  = { lp[k],  lp[k + 1] };
            v2f bv0 = { B[oB0], B[oB0 + bOff] };
            v2f bv1 = { B[oB1], B[oB1 + bOff] };
            acc0 = wmma_f32(av, bv0, acc0);
            acc1 = wmma_f32(av, bv1, acc1);
            oB0 += bStep; oB1 += bStep;
        }
    };

    // Last (partial) chunk: A is zero-padded in LDS, so clamp only B's k.
    auto compute_tail = [&](int buf, int kc, int kEnd) {
        const float* lp = &lsA[buf][ll * KPAD + 2 * lh];
        for (int k = 0; k < kEnd; k += 4) {
            const int kk  = kc + k + 2 * lh;
            const int k0c = min(kk, K - 1);
            const int k1c = min(kk + 1, K - 1);
            int i00, i01, i10, i11;
            if (TB) { i00 = bn0 * ldb + k0c; i01 = bn0 * ldb + k1c;
                      i10 = bn1 * ldb + k0c; i11 = bn1 * ldb + k1c; }
            else    { i00 = k0c * ldb + bn0; i01 = k1c * ldb + bn0;
                      i10 = k0c * ldb + bn1; i11 = k1c * ldb + bn1; }
            v2f av  = { lp[k],  lp[k + 1] };
            v2f bv0 = { B[i00], B[i01] };
            v2f bv1 = { B[i10], B[i11] };
            acc0 = wmma_f32(av, bv0, acc0);
            acc1 = wmma_f32(av, bv1, acc1);
        }
    };

    // Double-buffered pipeline over K chunks.
    fill(0, 0);
    wait_async();
    __syncthreads();
    int buf = 0;
    for (int kc = 0; kc < K; kc += KCH, buf ^= 1) {
        if (kc + KCH < K) fill(buf ^ 1, kc + KCH);     // stream next chunk
        if (active) {
            const int len = min(KCH, K - kc);
            if (len == KCH) compute_full(buf, kc);
            else            compute_tail(buf, kc, (len + 3) & ~3);
        }
        wait_async();
        __syncthreads();
    }

    if (!active) return;

    // C/D layout: VGPR v holds row m = m0 + v + 8*lh, col n = base + ll
    #pragma unroll
    for (int v = 0; v < 8; ++v) {
        const int m = m0 + v + 8 * lh;
        if (m >= M) continue;
        const float sc = scale ? scale[m] : 1.f;
        const float sh = scale ? shift[m] : 0.f;
        #pragma unroll
        for (int h = 0; h < 2; ++h) {
            const int n = n0 + 16 * h + ll;
            if (n >= N) continue;
            float val = (h == 0) ? acc0[v] : acc1[v];
            const long long ci = (long long)m * ldc + n;
            if (accum) val += C[ci];
            if (scale) {
                val = val * sc + sh;
                if (relu) val = fmaxf(val, 0.f);
            }
            C[ci] = val;
        }
    }
}

// out[b, col] = mean over r of in[b, r, col]
__global__ void rga_mean_rows(const float* __restrict__ in, float* __restrict__ out,
                              int rows, int cols, long long sIn, long long sOut)
{
    const int c = blockIdx.x * blockDim.x + threadIdx.x;
    const int b = blockIdx.y;
    if (c >= cols) return;
    const float* p = in + (long long)b * sIn + c;
    float acc = 0.f;
    for (int r = 0; r < rows; ++r) acc += p[(long long)r * cols];
    out[(long long)b * sOut + c] = acc / (float)rows;
}

// wys[b,s] = ws2_s * sum_d ws2_w[d]*t[b,d,s] + ws2_b ; xs = sigmoid(wys) * x
__global__ void rga_spatial_gate(const float* __restrict__ t,
                                 const float* __restrict__ w2,
                                 const float* __restrict__ s2,
                                 const float* __restrict__ b2,
                                 const float* __restrict__ x,
                                 float* __restrict__ xs)
{
    const int s = blockIdx.x * blockDim.x + threadIdx.x;
    const int b = blockIdx.y;
    if (s >= SS) return;
    const float* tp = t + (long long)b * DSN * SS + s;
    float acc = 0.f;
    for (int d = 0; d < DSN; ++d) acc += w2[d] * tp[(long long)d * SS];
    const float wys = acc * s2[0] + b2[0];
    const float g = 1.f / (1.f + expf(-wys));
    const long long base = (long long)b * CC * SS + s;
    for (int c = 0; c < CC; ++c) {
        const long long i = base + (long long)c * SS;
        xs[i] = g * x[i];
    }
}

// wyc[b,c] = wc2_s * sum_d wc2_w[d]*u[b,d,c] + wc2_b ; out = sigmoid(wyc) * xs
__global__ void rga_channel_gate(const float* __restrict__ u,
                                 const float* __restrict__ w2,
                                 const float* __restrict__ s2,
                                 const float* __restrict__ b2,
                                 const float* __restrict__ xs,
                                 float* __restrict__ out)
{
    const long long i = (long long)blockIdx.x * blockDim.x + threadIdx.x;
    const long long total = (long long)BB * CC * SS;
    if (i >= total) return;
    const long long bc = i / SS;
    const int c = (int)(bc % CC);
    const int b = (int)(bc / CC);
    const float* up = u + (long long)b * DCN * CC + c;
    float acc = 0.f;
    #pragma unroll
    for (int d = 0; d < DCN; ++d) acc += w2[d] * up[d * CC];
    const float wyc = acc * s2[0] + b2[0];
    const float g = 1.f / (1.f + expf(-wyc));
    out[i] = g * xs[i];
}

// ---------------------------------------------------------------------------
// Host side
// ---------------------------------------------------------------------------
static inline void launch_gemm(hipStream_t st, const float* A, const float* B, float* C,
                               int M, int N, int K, int lda, int ldb, int ldc,
                               long long sA, long long sB, long long sC,
                               int tA, int tB,
                               const float* scale, const float* shift,
                               int relu, int accum, int batch)
{
    dim3 grid((N + 32 * WAVES - 1) / (32 * WAVES), (M + 15) / 16, batch);
    const dim3 blk(WAVES * 32);
    if (tA == 0 && tB == 0)
        rga_wmma_gemm_f32<0, 0><<<grid, blk, 0, st>>>(A, B, C, M, N, K, lda, ldb, ldc,
                                                      sA, sB, sC, scale, shift, relu, accum);
    else if (tA == 0 && tB == 1)
        rga_wmma_gemm_f32<0, 1><<<grid, blk, 0, st>>>(A, B, C, M, N, K, lda, ldb, ldc,
                                                      sA, sB, sC, scale, shift, relu, accum);
    else if (tA == 1 && tB == 0)
        rga_wmma_gemm_f32<1, 0><<<grid, blk, 0, st>>>(A, B, C, M, N, K, lda, ldb, ldc,
                                                      sA, sB, sC, scale, shift, relu, accum);
    else
        rga_wmma_gemm_f32<1, 1><<<grid, blk, 0, st>>>(A, B, C, M, N, K, lda, ldb, ldc,
                                                      sA, sB, sC, scale, shift, relu, accum);
}

extern "C" void kernel_launch(void* const* d_in, const int* in_sizes, int n_in,
                              void* d_out, int out_size, void* d_ws, size_t ws_size,
                              hipStream_t stream)
{
    (void)in_sizes; (void)n_in; (void)out_size; (void)ws_size;
    // setup_inputs() dict order
    const float* x      = (const float*)d_in[0];
    const float* th_w   = (const float*)d_in[1];
    const float* ph_w   = (const float*)d_in[2];
    const float* ggs_w  = (const float*)d_in[3];   // [196, 1568]
    const float* gxs_w  = (const float*)d_in[4];
    const float* ws1_w  = (const float*)d_in[5];   // [49, 197]
    const float* ws2_w  = (const float*)d_in[6];   // [1, 49]
    const float* thc_w  = (const float*)d_in[7];   // [196, 784]
    const float* phc_w  = (const float*)d_in[8];
    const float* ggc_w  = (const float*)d_in[9];   // [32, 256]
    const float* gxc_w  = (const float*)d_in[10];
    const float* wc1_w  = (const float*)d_in[11];  // [8, 33]
    const float* wc2_w  = (const float*)d_in[12];  // [1, 8]
    const float* th_s   = (const float*)d_in[13];
    const float* th_b   = (const float*)d_in[14];
    const float* ph_s   = (const float*)d_in[15];
    const float* ph_b   = (const float*)d_in[16];
    const float* ggs_s  = (const float*)d_in[17];
    const float* ggs_b  = (const float*)d_in[18];
    const float* gxs_s  = (const float*)d_in[19];
    const float* gxs_b  = (const float*)d_in[20];
    const float* ws1_s  = (const float*)d_in[21];
    const float* ws1_b  = (const float*)d_in[22];
    const float* ws2_s  = (const float*)d_in[23];
    const float* ws2_b  = (const float*)d_in[24];
    const float* thc_s  = (const float*)d_in[25];
    const float* thc_b  = (const float*)d_in[26];
    const float* phc_s  = (const float*)d_in[27];
    const float* phc_b  = (const float*)d_in[28];
    const float* ggc_s  = (const float*)d_in[29];
    const float* ggc_b  = (const float*)d_in[30];
    const float* gxc_s  = (const float*)d_in[31];
    const float* gxc_b  = (const float*)d_in[32];
    const float* wc1_s  = (const float*)d_in[33];
    const float* wc1_b  = (const float*)d_in[34];
    const float* wc2_s  = (const float*)d_in[35];
    const float* wc2_b  = (const float*)d_in[36];

    float* out = (float*)d_out;

    // Workspace carving (floats). Reuse: thcout->theta, phcout->phi,
    // gxcout->gxsout (all 1,605,632 floats), Gc -> t buffer (t dead by then).
    float* ws = (float*)d_ws;
    const long long SZ_TP = (long long)BB * ICN * SS;   // 1,605,632 (== BB*ISN*CC)
    const long long SZ_P  = (long long)BB * ISN * ICN;  //   401,408
    const long long SZ_YS = (long long)BB * (1 + ISN) * SS;
    const long long SZ_T  = (long long)BB * DSN * SS;   // >= BB*CC*CC (Gc)
    const long long SZ_XS = (long long)BB * CC * SS;
    const long long SZ_YC = (long long)BB * (1 + ICN) * CC;

    float* theta  = ws;                 // [B,32,784]  ; later thcout [B,196,128]
    float* phi    = theta  + SZ_TP;     // [B,32,784]  ; later phcout
    float* gxsout = phi    + SZ_TP;     // [B,32,784]  ; later gxcout [B,196,128]
    float* P1     = gxsout + SZ_TP;     // [B,196,32]
    float* P2     = P1     + SZ_P;      // [B,196,32]
    float* ys     = P2     + SZ_P;      // [B,197,784]
    float* tbuf   = ys     + SZ_YS;     // [B,49,784]  ; later Gc [B,128,128]
    float* xs     = tbuf   + SZ_T;      // [B,128,784]
    float* yc     = xs     + SZ_XS;     // [B,33,128]
    float* ubuf   = yc     + SZ_YC;     // [B,8,128]

    float* thcout = theta;
    float* phcout = phi;
    float* gxcout = gxsout;
    float* Gc     = tbuf;

    // ---------------- spatial branch ----------------
    // theta/phi/gxs: [32,128] x [128,784] + affine + relu
    launch_gemm(stream, th_w,  x, theta,  ICN, SS, CC, CC, SS, SS,
                0, (long long)CC * SS, (long long)ICN * SS, 0, 0, th_s,  th_b,  1, 0, BB);
    launch_gemm(stream, ph_w,  x, phi,    ICN, SS, CC, CC, SS, SS,
                0, (long long)CC * SS, (long long)ICN * SS, 0, 0, ph_s,  ph_b,  1, 0, BB);
    launch_gemm(stream, gxs_w, x, gxsout, ICN, SS, CC, CC, SS, SS,
                0, (long long)CC * SS, (long long)ICN * SS, 0, 0, gxs_s, gxs_b, 1, 0, BB);

    // ys row 0 = mean over 32 rows of gxsout
    rga_mean_rows<<<dim3((SS + 255) / 256, BB), 256, 0, stream>>>(
        gxsout, ys, ICN, SS, (long long)ICN * SS, (long long)(1 + ISN) * SS);

    // Factorized Gs_joint: P1 = w1*phi^T, P2 = w2*theta^T  ([196,784]x[784,32])
    launch_gemm(stream, ggs_w,      phi,   P1, ISN, ICN, SS, 2 * SS, SS, ICN,
                0, (long long)ICN * SS, (long long)ISN * ICN, 0, 1, nullptr, nullptr, 0, 0, BB);
    launch_gemm(stream, ggs_w + SS, theta, P2, ISN, ICN, SS, 2 * SS, SS, ICN,
                0, (long long)ICN * SS, (long long)ISN * ICN, 0, 1, nullptr, nullptr, 0, 0, BB);
    // ys rows 1..196 = relu(bn(P1*theta + P2*phi))
    launch_gemm(stream, P1, theta, ys + SS, ISN, SS, ICN, ICN, SS, SS,
                (long long)ISN * ICN, (long long)ICN * SS, (long long)(1 + ISN) * SS,
                0, 0, nullptr, nullptr, 0, 0, BB);
    launch_gemm(stream, P2, phi,   ys + SS, ISN, SS, ICN, ICN, SS, SS,
                (long long)ISN * ICN, (long long)ICN * SS, (long long)(1 + ISN) * SS,
                0, 0, ggs_s, ggs_b, 1, 1, BB);

    // t = relu(bn(ws1 * ys)) : [49,197] x [197,784]
    launch_gemm(stream, ws1_w, ys, tbuf, DSN, SS, 1 + ISN, 1 + ISN, SS, SS,
                0, (long long)(1 + ISN) * SS, (long long)DSN * SS,
                0, 0, ws1_s, ws1_b, 1, 0, BB);

    // xs = sigmoid(ws2 path) * x
    rga_spatial_gate<<<dim3((SS + 255) / 256, BB), 256, 0, stream>>>(
        tbuf, ws2_w, ws2_s, ws2_b, x, xs);

    // ---------------- channel branch ----------------
    // thc/phc/gxc: [196,784] x xs^T(784x128)  (transB) + affine + relu
    launch_gemm(stream, thc_w, xs, thcout, ISN, CC, SS, SS, SS, CC,
                0, (long long)CC * SS, (long long)ISN * CC, 0, 1, thc_s, thc_b, 1, 0, BB);
    launch_gemm(stream, phc_w, xs, phcout, ISN, CC, SS, SS, SS, CC,
                0, (long long)CC * SS, (long long)ISN * CC, 0, 1, phc_s, phc_b, 1, 0, BB);
    launch_gemm(stream, gxc_w, xs, gxcout, ISN, CC, SS, SS, SS, CC,
                0, (long long)CC * SS, (long long)ISN * CC, 0, 1, gxc_s, gxc_b, 1, 0, BB);

    // yc row 0 = mean over 196 rows of gxcout
    rga_mean_rows<<<dim3((CC + 255) / 256, BB), 256, 0, stream>>>(
        gxcout, yc, ISN, CC, (long long)ISN * CC, (long long)(1 + ICN) * CC);

    // Gc = thcout^T * phcout : [128,196]^T-access x [196,128]
    launch_gemm(stream, thcout, phcout, Gc, CC, CC, ISN, CC, CC, CC,
                (long long)ISN * CC, (long long)ISN * CC, (long long)CC * CC,
                1, 0, nullptr, nullptr, 0, 0, BB);

    // yc rows 1..32 = relu(bn(w1*Gc^T + w2*Gc)) ; ggc_w is [32, 256]
    launch_gemm(stream, ggc_w,      Gc, yc + CC, ICN, CC, CC, 2 * CC, CC, CC,
                0, (long long)CC * CC, (long long)(1 + ICN) * CC,
                0, 1, nullptr, nullptr, 0, 0, BB);
    launch_gemm(stream, ggc_w + CC, Gc, yc + CC, ICN, CC, CC, 2 * CC, CC, CC,
                0, (long long)CC * CC, (long long)(1 + ICN) * CC,
                0, 0, ggc_s, ggc_b, 1, 1, BB);

    // u = relu(bn(wc1 * yc)) : [8,33] x [33,128]
    launch_gemm(stream, wc1_w, yc, ubuf, DCN, CC, 1 + ICN, 1 + ICN, CC, CC,
                0, (long long)(1 + ICN) * CC, (long long)DCN * CC,
                0, 0, wc1_s, wc1_b, 1, 0, BB);

    // out = sigmoid(wc2 path) * xs
    const long long total = (long long)BB * CC * SS;
    rga_channel_gate<<<dim3((unsigned)((total + 255) / 256)), 256, 0, stream>>>(
        ubuf, wc2_w, wc2_s, wc2_b, xs, out);
}